// KorniaDA_84688165143260
// MI455X (gfx1250) — compile-verified
//
#include <hip/hip_runtime.h>
#include <stdint.h>
#include <math.h>

#define OUTD    96
#define HW      224
#define NPIX    (OUTD * OUTD)          // 9216
#define THREADS 512                     // 16 wave32 waves
#define PPT     (NPIX / THREADS)        // 18 pixels per thread
#define IMG_ELEMS (3 * HW * HW)         // 150528

// ---------------- Threefry-2x32 (JAX-compatible, classic path) ----------------
__host__ __device__ inline uint32_t rotl32(uint32_t x, int n) {
  return (x << n) | (x >> (32 - n));
}

__host__ __device__ inline void threefry2x32(uint32_t k0, uint32_t k1,
                                             uint32_t& x0, uint32_t& x1) {
  const uint32_t ks2 = k0 ^ k1 ^ 0x1BD11BDAu;
  x0 += k0; x1 += k1;
#define TF_RND(r) { x0 += x1; x1 = rotl32(x1, (r)); x1 ^= x0; }
  TF_RND(13) TF_RND(15) TF_RND(26) TF_RND(6)
  x0 += k1;  x1 += ks2 + 1u;
  TF_RND(17) TF_RND(29) TF_RND(16) TF_RND(24)
  x0 += ks2; x1 += k0 + 2u;
  TF_RND(13) TF_RND(15) TF_RND(26) TF_RND(6)
  x0 += k0;  x1 += k1 + 3u;
  TF_RND(17) TF_RND(29) TF_RND(16) TF_RND(24)
  x0 += k1;  x1 += ks2 + 4u;
  TF_RND(13) TF_RND(15) TF_RND(26) TF_RND(6)
  x0 += ks2; x1 += k0 + 5u;
#undef TF_RND
}

struct SubKeys { uint32_t k[20]; };     // 10 (k0,k1) pairs from jax.random.split

__device__ inline float u01_from_bits(uint32_t bits) {
  // JAX uniform: bitcast((bits>>9)|0x3f800000) - 1.0  -> [0,1)
  return __uint_as_float((bits >> 9) | 0x3f800000u) - 1.0f;
}

__device__ inline float clip01(float v) { return fminf(fmaxf(v, 0.0f), 1.0f); }

// CDNA5 async LDS->global streaming (guarded; fallback = plain stores)
#if __has_builtin(__builtin_amdgcn_global_store_async_from_lds_b128) && \
    __has_builtin(__builtin_amdgcn_s_wait_asynccnt)
#define HAVE_ASYNC_LDS 1
// Parameter types per hipcc diagnostic: pointee is gcc-style int vector_size(16),
// global side in addrspace(1) ("__device__"), LDS side in addrspace(3).
typedef int v4i_b128 __attribute__((vector_size(16)));
typedef __attribute__((address_space(1))) v4i_b128* g_b128_p;
typedef __attribute__((address_space(3))) v4i_b128* l_b128_p;
#else
#define HAVE_ASYNC_LDS 0
#endif

// ---------------- One workgroup per image ----------------
__global__ __launch_bounds__(THREADS)
void kornia_aug_kernel(const float* __restrict__ in, float* __restrict__ out,
                       SubKeys sk, int half /* = B/2 */) {
  extern __shared__ float s_img[];      // 3 * NPIX floats (108 KB, CDNA5 LDS)
  __shared__ float s_raw[10];
  __shared__ float s_p[9];
  __shared__ float s_red[THREADS];

  const int b   = blockIdx.x;
  const int tid = threadIdx.x;

  // --- per-image random parameters (threads 0..9, one threefry block each) ---
  if (tid < 10) {
    uint32_t x0, x1; int hi;
    if (b < half) { x0 = (uint32_t)b;          x1 = (uint32_t)(b + half); hi = 0; }
    else          { x0 = (uint32_t)(b - half); x1 = (uint32_t)b;          hi = 1; }
    threefry2x32(sk.k[2 * tid], sk.k[2 * tid + 1], x0, x1);
    s_raw[tid] = u01_from_bits(hi ? x1 : x0);
  }
  __syncthreads();

  if (tid == 0) {
    float area = 224.0f * 224.0f * (s_raw[1] * 0.92f + 0.08f);
    const float lo = -0.28768207f, hv = 0.28768207f;         // log(3/4), log(4/3)
    float ratio = expf(s_raw[2] * (hv - lo) + lo);
    float cw = fminf(fmaxf(sqrtf(area * ratio), 1.0f), 224.0f);
    float ch = fminf(fmaxf(sqrtf(area / ratio), 1.0f), 224.0f);
    int appm = s_raw[5] < 0.8f;
    s_p[0] = (s_raw[0] < 0.5f) ? 1.0f : 0.0f;                // flip
    s_p[1] = s_raw[4] * (224.0f - ch);                       // y0
    s_p[2] = s_raw[3] * (224.0f - cw);                       // x0
    s_p[3] = (ch - 1.0f) / 95.0f;                            // sy
    s_p[4] = (cw - 1.0f) / 95.0f;                            // sx
    s_p[5] = appm ? (s_raw[6] + 0.5f) : 1.0f;                // brightness
    s_p[6] = appm ? (s_raw[7] + 0.5f) : 1.0f;                // contrast
    s_p[7] = appm ? (s_raw[8] + 0.5f) : 1.0f;                // saturation
    s_p[8] = appm ? (s_raw[9] * 0.2f - 0.1f) : 0.0f;         // hue
  }
  __syncthreads();

  const int   flip = s_p[0] != 0.0f;
  const float y0p = s_p[1], x0p = s_p[2], sy = s_p[3], sx = s_p[4], bri = s_p[5];
  const float* __restrict__ imgb = in + (size_t)b * IMG_ELEMS;

  // --- Phase 1: flip + bilinear crop + brightness -> LDS; gray partial sum ---
  float lsum = 0.0f;
#pragma unroll
  for (int kk = 0; kk < PPT; ++kk) {
    int p = tid + kk * THREADS;
    int j = p / OUTD, i = p - j * OUTD;
    float ys = y0p + (float)j * sy;
    float xs = x0p + (float)i * sx;
    float yf = floorf(ys), xf = floorf(xs);
    float wy = ys - yf,   wx = xs - xf;
    int yi0 = min(max((int)yf, 0), HW - 1);
    int yi1 = min(yi0 + 1, HW - 1);
    int xi0 = min(max((int)xf, 0), HW - 1);
    int xi1 = min(xi0 + 1, HW - 1);
    if (flip) { xi0 = HW - 1 - xi0; xi1 = HW - 1 - xi1; }

    const float* r0 = imgb + yi0 * HW;
    const float* r1 = imgb + yi1 * HW;
    // CDNA5 global_prefetch_b8: warm channel 1/2 lines before their gathers
    __builtin_prefetch(r0 + HW * HW + xi0, 0, 3);
    __builtin_prefetch(r0 + 2 * HW * HW + xi0, 0, 3);

    float rgb[3];
#pragma unroll
    for (int c = 0; c < 3; ++c) {
      const float* rc0 = r0 + c * HW * HW;
      const float* rc1 = r1 + c * HW * HW;
      float a0 = rc0[xi0] * (1.0f - wy) + rc1[xi0] * wy;     // rows mix (y)
      float a1 = rc0[xi1] * (1.0f - wy) + rc1[xi1] * wy;
      float v  = a0 * (1.0f - wx) + a1 * wx;                 // cols mix (x)
      v = clip01(v * bri);
      rgb[c] = v;
      s_img[c * NPIX + p] = v;
    }
    lsum += 0.299f * rgb[0] + 0.587f * rgb[1] + 0.114f * rgb[2];
  }

  // --- deterministic tree reduction for per-image gray mean ---
  s_red[tid] = lsum;
  __syncthreads();
  for (int off = THREADS / 2; off > 0; off >>= 1) {
    if (tid < off) s_red[tid] += s_red[tid + off];
    __syncthreads();
  }
  const float mean = s_red[0] / (float)NPIX;

  // --- Phase 2: contrast, saturation, HSV hue, clip, normalize ---
  const float con = s_p[6], sat = s_p[7], hue = s_p[8];
  float* __restrict__ outb = out + (size_t)b * 3 * NPIX;
#pragma unroll
  for (int kk = 0; kk < PPT; ++kk) {
    int p = tid + kk * THREADS;
    float r  = s_img[p];
    float g  = s_img[NPIX + p];
    float bl = s_img[2 * NPIX + p];
    float gray = 0.299f * r + 0.587f * g + 0.114f * bl;

    r  = clip01(r  * con + mean * (1.0f - con));
    g  = clip01(g  * con + mean * (1.0f - con));
    bl = clip01(bl * con + mean * (1.0f - con));
    r  = clip01(r  * sat + gray * (1.0f - sat));
    g  = clip01(g  * sat + gray * (1.0f - sat));
    bl = clip01(bl * sat + gray * (1.0f - sat));

    // RGB -> HSV
    float maxc = fmaxf(r, fmaxf(g, bl));
    float minc = fminf(r, fminf(g, bl));
    float v = maxc;
    float d = maxc - minc;
    float s = d / fmaxf(maxc, 1e-8f);
    float sd = fmaxf(d, 1e-8f);
    float rc = (maxc - r) / sd;
    float gc = (maxc - g) / sd;
    float bc = (maxc - bl) / sd;
    float h = (maxc == r) ? (bc - gc)
            : ((maxc == g) ? (2.0f + rc - bc) : (4.0f + gc - rc));
    h = h / 6.0f;
    h = h - floorf(h);                      // python-style mod 1
    h = (d <= 1e-8f) ? 0.0f : h;
    h = h + hue;
    h = h - floorf(h);

    // HSV -> RGB
    float hf = h * 6.0f;
    float fi = floorf(hf);
    float f  = hf - fi;
    int ii = ((int)fi) % 6;
    float pv = v * (1.0f - s);
    float qv = v * (1.0f - s * f);
    float tv = v * (1.0f - s * (1.0f - f));
    float ro, go, bo;
    switch (ii) {
      case 0:  ro = v;  go = tv; bo = pv; break;
      case 1:  ro = qv; go = v;  bo = pv; break;
      case 2:  ro = pv; go = v;  bo = tv; break;
      case 3:  ro = pv; go = qv; bo = v;  break;
      case 4:  ro = tv; go = pv; bo = v;  break;
      default: ro = v;  go = pv; bo = qv; break;
    }
    ro = (clip01(ro) - 0.5f) / 0.5f;
    go = (clip01(go) - 0.5f) / 0.5f;
    bo = (clip01(bo) - 0.5f) / 0.5f;

#if HAVE_ASYNC_LDS
    // write final result back into LDS in place (same slots, same thread)
    s_img[p]            = ro;
    s_img[NPIX + p]     = go;
    s_img[2 * NPIX + p] = bo;
#else
    outb[p]            = ro;
    outb[NPIX + p]     = go;
    outb[2 * NPIX + p] = bo;
#endif
  }

#if HAVE_ASYNC_LDS
  // --- CDNA5 async drain: LDS -> global, B128 per lane, tracked on ASYNCcnt ---
  __syncthreads();
  const int nchunks = (3 * NPIX) / 4;     // 6912 x 16B chunks
  for (int q = tid; q < nchunks; q += THREADS) {
    __builtin_amdgcn_global_store_async_from_lds_b128(
        (g_b128_p)(outb + 4 * q), (l_b128_p)(s_img + 4 * q), 0, 0);
  }
  __builtin_amdgcn_s_wait_asynccnt(0);
#endif
}

extern "C" void kernel_launch(void* const* d_in, const int* in_sizes, int n_in,
                              void* d_out, int out_size, void* d_ws, size_t ws_size,
                              hipStream_t stream) {
  (void)n_in; (void)d_ws; (void)ws_size; (void)out_size;
  const float* x = (const float*)d_in[0];
  float* out = (float*)d_out;

  const int B = in_sizes[0] / IMG_ELEMS;      // 128
  const int half = B >> 1;

  // jax.random.key(42) = (0,42); split(key,10): threefry over iota(20),
  // reshape (10,2). Pure, deterministic host computation (graph-capture safe).
  SubKeys sk;
  uint32_t outv[20];
  for (int pidx = 0; pidx < 10; ++pidx) {
    uint32_t a = (uint32_t)pidx, c = (uint32_t)(pidx + 10);
    threefry2x32(0u, 42u, a, c);
    outv[pidx] = a; outv[pidx + 10] = c;
  }
  for (int i = 0; i < 20; ++i) sk.k[i] = outv[i];

  const size_t shmem = (size_t)3 * NPIX * sizeof(float);   // 110592 B < 320 KB
  kornia_aug_kernel<<<B, THREADS, shmem, stream>>>(x, out, sk, half);
}